// MultiPatternModel_34256659153280
// MI455X (gfx1250) — compile-verified
//
#include <hip/hip_runtime.h>
#include <cstdint>
#include <cmath>

typedef __attribute__((ext_vector_type(2))) float v2f;
typedef __attribute__((ext_vector_type(8))) float v8f;

#define B_N 8192
#define D_N 2048
#define H_N 2048
#define Z_N 256
#define E_N 5
#define C_N 100

#define TM 64
#define TN 64
#define TK 32
#define SA 36     // padded LDS stride for A tile (floats)
#define SBP 160   // LDS stride per B k-pair row (128 data + 32 pad; ==32 mod 64 -> conflict-free)

// ---------------------------------------------------------------------------
// Core 64x64 macro-tile GEMM: 8 waves, each wave owns a 16x32 slice (two
// 16x16 WMMA accumulators). A/B staged via LDS in the exact CDNA5 WMMA
// f32 16x16x4 operand layouts (ISA 7.12.2):
//   A: lane l -> m = l%16, VGPR0/1 = A[m][2*(l/16) + 0/1]   (row-major, contiguous)
//   B: lane l -> n = l%16, VGPR0/1 = B[2*(l/16) + 0/1][n]   (k-pair interleaved in LDS
//      so both VGPRs come from ONE ds_load_b64 -> no register-shuffle before WMMA)
// ---------------------------------------------------------------------------
template<bool GATHER>
__device__ inline void gemm64(const float* __restrict__ A, int lda,
                              const float* __restrict__ Bm, int ldb,
                              int K, int rowBase, int rowLimit,
                              const int* __restrict__ gather,
                              int ncolBase,
                              float* sA, float* sB,
                              v8f& c0, v8f& c1)
{
  const int t    = threadIdx.x;
  const int lane = t & 31;
  const int wave = t >> 5;
  const int wm   = (wave & 3) * 16;
  const int wn   = (wave >> 2) * 32;
  const int mrow = lane & 15;
  const int khalf = (lane >> 4) << 1;       // 0 or 2

  for (int k0 = 0; k0 < K; k0 += TK) {
    __syncthreads();
#pragma unroll
    for (int i = 0; i < 2; ++i) {
      int li = t * 2 + i;                   // 0..511
      // A tile: 64 rows x 32 cols (512 float4), row-major with padded stride
      int ar = li >> 3;
      int ac = (li & 7) << 2;
      int lp = rowBase + ar;
      int arow;
      if (GATHER) {
        int cl = lp < rowLimit ? lp : (rowLimit - 1);
        arow = gather[cl];
      } else {
        arow = lp < rowLimit ? lp : (rowLimit - 1);
      }
      const float4 av = *(const float4*)(A + (size_t)arow * lda + k0 + ac);
      *(float4*)(sA + ar * SA + ac) = av;
      // B tile: 32 k-rows x 64 cols, stored k-pair interleaved:
      //   sB[pair*SBP + n*2 + (k&1)] = B[k0 + pair*2 + (k&1)][ncolBase + n]
      int pair = li >> 5;                   // 0..15
      int nq   = li & 31;                   // n pair index, n = nq*2
      const float* gb = Bm + (size_t)(k0 + pair * 2) * ldb + ncolBase + nq * 2;
      const float2 brow0 = *(const float2*)(gb);
      const float2 brow1 = *(const float2*)(gb + ldb);
      float4 q; q.x = brow0.x; q.y = brow1.x; q.z = brow0.y; q.w = brow1.y;
      *(float4*)(sB + pair * SBP + nq * 4) = q;
    }
    if (k0 + TK < K) {
      // hint the next B K-slab toward the caches (global_prefetch_b8)
      __builtin_prefetch(Bm + (size_t)(k0 + TK + (t >> 3)) * ldb + ncolBase + ((t & 7) << 3), 0, 1);
    }
    __syncthreads();

    const float* pa = sA + (wm + mrow) * SA + khalf;
    const float* pb = sB + (khalf >> 1) * SBP + (wn + (lane & 15)) * 2;
#pragma unroll
    for (int k4 = 0; k4 < TK; k4 += 4) {
      v2f a  = *(const v2f*)(pa);           // {A[m][k], A[m][k+1]}
      v2f b0 = *(const v2f*)(pb);           // {B[k][n],    B[k+1][n]}
      v2f b1 = *(const v2f*)(pb + 32);      // {B[k][n+16], B[k+1][n+16]}
      c0 = __builtin_amdgcn_wmma_f32_16x16x4_f32(false, a, false, b0, (short)0, c0, false, false);
      c1 = __builtin_amdgcn_wmma_f32_16x16x4_f32(false, a, false, b1, (short)0, c1, false, false);
      pa += 4;
      pb += 2 * SBP;                        // k advances by 4 -> 2 k-pair rows
    }
  }
}

__device__ inline float wave_reduce_sum(float v) {
#pragma unroll
  for (int off = 16; off; off >>= 1) v += __shfl_xor(v, off, 32);
  return v;
}

// ---------------------------------------------------------------------------
// K1: h = relu(BN(feats @ pW1 + pb1))            [B,H], K=D
// ---------------------------------------------------------------------------
__global__ __launch_bounds__(256)
void k1_gemm_bn_relu(const float* __restrict__ feats, const float* __restrict__ pW1,
                     const float* __restrict__ pb1, const float* __restrict__ pg,
                     const float* __restrict__ pbt, const float* __restrict__ pm,
                     const float* __restrict__ pv, float* __restrict__ hbuf)
{
  __shared__ float sA[TM * SA];
  __shared__ float sB[(TK / 2) * SBP];
  v8f c0 = {}; v8f c1 = {};
  const int tileN = blockIdx.x * TN;
  const int tileM = blockIdx.y * TM;
  gemm64<false>(feats, D_N, pW1, H_N, D_N, tileM, B_N, nullptr, tileN, sA, sB, c0, c1);

  const int lane = threadIdx.x & 31;
  const int wave = threadIdx.x >> 5;
  const int wm = (wave & 3) * 16;
  const int wn = (wave >> 2) * 32;
#pragma unroll
  for (int v = 0; v < 8; ++v) {
    int r = tileM + wm + v + ((lane >> 4) << 3);
#pragma unroll
    for (int s = 0; s < 2; ++s) {
      int c = tileN + wn + (lane & 15) + s * 16;
      float acc = s ? c1[v] : c0[v];
      float val = (acc + pb1[c] - pm[c]) * (pg[c] * rsqrtf(pv[c] + 1e-5f)) + pbt[c];
      hbuf[(size_t)r * H_N + c] = val > 0.f ? val : 0.f;
    }
  }
}

// ---------------------------------------------------------------------------
// K2: zpraw = h @ pW2 + pb2                      [B,Z], K=H
// ---------------------------------------------------------------------------
__global__ __launch_bounds__(256)
void k2_gemm_bias(const float* __restrict__ hbuf, const float* __restrict__ pW2,
                  const float* __restrict__ pb2, float* __restrict__ zp)
{
  __shared__ float sA[TM * SA];
  __shared__ float sB[(TK / 2) * SBP];
  v8f c0 = {}; v8f c1 = {};
  const int tileN = blockIdx.x * TN;
  const int tileM = blockIdx.y * TM;
  gemm64<false>(hbuf, H_N, pW2, Z_N, H_N, tileM, B_N, nullptr, tileN, sA, sB, c0, c1);

  const int lane = threadIdx.x & 31;
  const int wave = threadIdx.x >> 5;
  const int wm = (wave & 3) * 16;
  const int wn = (wave >> 2) * 32;
#pragma unroll
  for (int v = 0; v < 8; ++v) {
    int r = tileM + wm + v + ((lane >> 4) << 3);
#pragma unroll
    for (int s = 0; s < 2; ++s) {
      int c = tileN + wn + (lane & 15) + s * 16;
      float acc = s ? c1[v] : c0[v];
      zp[(size_t)r * Z_N + c] = acc + pb2[c];
    }
  }
}

// ---------------------------------------------------------------------------
// K3: in-place row L2-normalize of zp, pattern logits, argmax route.
// One wave per row (Z=256 -> 8 elems per lane).
// ---------------------------------------------------------------------------
__global__ __launch_bounds__(256)
void k3_norm_route(float* __restrict__ zp, const float* __restrict__ pProto,
                   float* __restrict__ logits_pattern, int* __restrict__ route)
{
  const int wave = threadIdx.x >> 5;
  const int lane = threadIdx.x & 31;
  const int row = blockIdx.x * 8 + wave;
  float x[8];
  float s = 0.f;
#pragma unroll
  for (int j = 0; j < 8; ++j) {
    x[j] = zp[(size_t)row * Z_N + lane + 32 * j];
    s += x[j] * x[j];
  }
  s = wave_reduce_sum(s);
  float inv = 1.f / fmaxf(sqrtf(s), 1e-12f);
#pragma unroll
  for (int j = 0; j < 8; ++j) {
    x[j] *= inv;
    zp[(size_t)row * Z_N + lane + 32 * j] = x[j];
  }
  float best = -1e30f; int bidx = 0;
#pragma unroll
  for (int e = 0; e < E_N; ++e) {
    float p = 0.f;
#pragma unroll
    for (int j = 0; j < 8; ++j) p += x[j] * pProto[(size_t)e * Z_N + lane + 32 * j];
    p = wave_reduce_sum(p);
    if (lane == 0) logits_pattern[(size_t)row * E_N + e] = p;
    if (p > best) { best = p; bidx = e; }   // first-max wins (ascending e)
  }
  if (lane == 0) route[row] = bidx;
}

// ---------------------------------------------------------------------------
// Routing bookkeeping: counts -> offsets -> gather list.
// ---------------------------------------------------------------------------
__global__ void k0_zero(int* __restrict__ counts) {
  if (threadIdx.x < 8) counts[threadIdx.x] = 0;
}
__global__ void k4_count(const int* __restrict__ route, int* __restrict__ counts) {
  int i = blockIdx.x * blockDim.x + threadIdx.x;
  if (i < B_N) atomicAdd(&counts[route[i]], 1);
}
__global__ void k4_prefix(const int* __restrict__ counts, int* __restrict__ offsets,
                          int* __restrict__ cursor) {
  if (threadIdx.x == 0) {
    int o = 0;
    for (int e = 0; e < E_N; ++e) { offsets[e] = o; cursor[e] = o; o += counts[e]; }
  }
}
__global__ void k4_fill(const int* __restrict__ route, int* __restrict__ cursor,
                        int* __restrict__ glist) {
  int i = blockIdx.x * blockDim.x + threadIdx.x;
  if (i < B_N) {
    int e = route[i];
    int p = atomicAdd(&cursor[e], 1);
    glist[p] = i;
  }
}

// ---------------------------------------------------------------------------
// K5: gathered expert GEMM1: h2g[pos] = relu(BN_e(zp[glist[pos]] @ cW1[e] + cb1[e]))
// grid: (H/64, B/64, E); tiles past counts[e] exit early.
// ---------------------------------------------------------------------------
__global__ __launch_bounds__(256)
void k5_expert1(const float* __restrict__ zp, const int* __restrict__ glist,
                const int* __restrict__ counts, const int* __restrict__ offsets,
                const float* __restrict__ cW1, const float* __restrict__ cb1,
                const float* __restrict__ cg, const float* __restrict__ cbt,
                const float* __restrict__ cm, const float* __restrict__ cv,
                float* __restrict__ h2g)
{
  const int e = blockIdx.z;
  const int cnt = counts[e];
  const int tRow = blockIdx.y * TM;
  if (tRow >= cnt) return;
  const int goff = offsets[e];
  __shared__ float sA[TM * SA];
  __shared__ float sB[(TK / 2) * SBP];
  v8f c0 = {}; v8f c1 = {};
  const int tileN = blockIdx.x * TN;
  gemm64<true>(zp, Z_N, cW1 + (size_t)e * Z_N * H_N, H_N, Z_N,
               tRow, cnt, glist + goff, tileN, sA, sB, c0, c1);

  const int lane = threadIdx.x & 31;
  const int wave = threadIdx.x >> 5;
  const int wm = (wave & 3) * 16;
  const int wn = (wave >> 2) * 32;
#pragma unroll
  for (int v = 0; v < 8; ++v) {
    int lr = tRow + wm + v + ((lane >> 4) << 3);
    if (lr < cnt) {
#pragma unroll
      for (int s = 0; s < 2; ++s) {
        int c = tileN + wn + (lane & 15) + s * 16;
        float acc = s ? c1[v] : c0[v];
        float val = (acc + cb1[e * H_N + c] - cm[e * H_N + c]) *
                    (cg[e * H_N + c] * rsqrtf(cv[e * H_N + c] + 1e-5f)) + cbt[e * H_N + c];
        h2g[(size_t)(goff + lr) * H_N + c] = val > 0.f ? val : 0.f;
      }
    }
  }
}

// ---------------------------------------------------------------------------
// K6: expert GEMM2: z2raw[pos] = h2g[pos] @ cW2[e] + cb2[e]   K=H
// ---------------------------------------------------------------------------
__global__ __launch_bounds__(256)
void k6_expert2(const float* __restrict__ h2g, const int* __restrict__ counts,
                const int* __restrict__ offsets, const float* __restrict__ cW2,
                const float* __restrict__ cb2, float* __restrict__ z2raw)
{
  const int e = blockIdx.z;
  const int cnt = counts[e];
  const int tRow = blockIdx.y * TM;
  if (tRow >= cnt) return;
  const int goff = offsets[e];
  __shared__ float sA[TM * SA];
  __shared__ float sB[(TK / 2) * SBP];
  v8f c0 = {}; v8f c1 = {};
  const int tileN = blockIdx.x * TN;
  gemm64<false>(h2g + (size_t)goff * H_N, H_N, cW2 + (size_t)e * H_N * Z_N, Z_N, H_N,
                tRow, cnt, nullptr, tileN, sA, sB, c0, c1);

  const int lane = threadIdx.x & 31;
  const int wave = threadIdx.x >> 5;
  const int wm = (wave & 3) * 16;
  const int wn = (wave >> 2) * 32;
#pragma unroll
  for (int v = 0; v < 8; ++v) {
    int lr = tRow + wm + v + ((lane >> 4) << 3);
    if (lr < cnt) {
#pragma unroll
      for (int s = 0; s < 2; ++s) {
        int c = tileN + wn + (lane & 15) + s * 16;
        float acc = s ? c1[v] : c0[v];
        z2raw[(size_t)(goff + lr) * Z_N + c] = acc + cb2[e * Z_N + c];
      }
    }
  }
}

// ---------------------------------------------------------------------------
// K7: normalize z2 rows, class logits vs cProto[e], scatter to original rows.
// One wave per gathered row.
// ---------------------------------------------------------------------------
__global__ __launch_bounds__(256)
void k7_final(const float* __restrict__ z2raw, const int* __restrict__ glist,
              const int* __restrict__ counts, const int* __restrict__ offsets,
              const float* __restrict__ cProto, float* __restrict__ proj_class,
              float* __restrict__ logits_class)
{
  const int wave = threadIdx.x >> 5;
  const int lane = threadIdx.x & 31;
  const int pos = blockIdx.x * 8 + wave;
  int e = 0;
#pragma unroll
  for (int k = 0; k < E_N; ++k)
    if (pos >= offsets[k] + counts[k]) e = k + 1;
  if (e >= E_N) e = E_N - 1;
  const int orig = glist[pos];

  float x[8];
  float s = 0.f;
#pragma unroll
  for (int j = 0; j < 8; ++j) {
    x[j] = z2raw[(size_t)pos * Z_N + lane + 32 * j];
    s += x[j] * x[j];
  }
  s = wave_reduce_sum(s);
  float inv = 1.f / fmaxf(sqrtf(s), 1e-12f);
#pragma unroll
  for (int j = 0; j < 8; ++j) {
    x[j] *= inv;
    proj_class[(size_t)orig * Z_N + lane + 32 * j] = x[j];
  }
  const float* P = cProto + (size_t)e * C_N * Z_N;
  for (int c = 0; c < C_N; ++c) {
    float p = 0.f;
#pragma unroll
    for (int j = 0; j < 8; ++j) p += x[j] * P[(size_t)c * Z_N + lane + 32 * j];
    p = wave_reduce_sum(p);
    if (lane == 0) logits_class[(size_t)orig * C_N + c] = p;
  }
}

// ---------------------------------------------------------------------------
extern "C" void kernel_launch(void* const* d_in, const int* in_sizes, int n_in,
                              void* d_out, int out_size, void* d_ws, size_t ws_size,
                              hipStream_t stream) {
  (void)in_sizes; (void)n_in; (void)out_size; (void)ws_size;
  const float* feats  = (const float*)d_in[0];
  const float* pW1    = (const float*)d_in[1];
  const float* pb1    = (const float*)d_in[2];
  const float* pg     = (const float*)d_in[3];
  const float* pbt    = (const float*)d_in[4];
  const float* pm     = (const float*)d_in[5];
  const float* pv     = (const float*)d_in[6];
  const float* pW2    = (const float*)d_in[7];
  const float* pb2    = (const float*)d_in[8];
  const float* pProto = (const float*)d_in[9];
  const float* cW1    = (const float*)d_in[10];
  const float* cb1    = (const float*)d_in[11];
  const float* cg     = (const float*)d_in[12];
  const float* cbt    = (const float*)d_in[13];
  const float* cm     = (const float*)d_in[14];
  const float* cv     = (const float*)d_in[15];
  const float* cW2    = (const float*)d_in[16];
  const float* cb2    = (const float*)d_in[17];
  const float* cProto = (const float*)d_in[18];

  float* out = (float*)d_out;
  float* logits_pattern = out;                                   // [B,E]
  float* zp             = logits_pattern + (size_t)B_N * E_N;    // [B,Z]
  float* logits_class   = zp + (size_t)B_N * Z_N;                // [B,C]
  float* proj_class     = logits_class + (size_t)B_N * C_N;      // [B,Z]

  float* hbuf  = (float*)d_ws;                   // [B,H] (reused as h2g after K2)
  float* z2raw = hbuf + (size_t)B_N * H_N;       // [B,Z]
  int* route   = (int*)(z2raw + (size_t)B_N * Z_N);
  int* glist   = route + B_N;
  int* counts  = glist + B_N;
  int* offsets = counts + 8;
  int* cursor  = offsets + 8;

  k0_zero<<<1, 32, 0, stream>>>(counts);
  k1_gemm_bn_relu<<<dim3(H_N / TN, B_N / TM), 256, 0, stream>>>(
      feats, pW1, pb1, pg, pbt, pm, pv, hbuf);
  k2_gemm_bias<<<dim3(Z_N / TN, B_N / TM), 256, 0, stream>>>(hbuf, pW2, pb2, zp);
  k3_norm_route<<<B_N / 8, 256, 0, stream>>>(zp, pProto, logits_pattern, route);
  k4_count<<<B_N / 256, 256, 0, stream>>>(route, counts);
  k4_prefix<<<1, 32, 0, stream>>>(counts, offsets, cursor);
  k4_fill<<<B_N / 256, 256, 0, stream>>>(route, cursor, glist);
  k5_expert1<<<dim3(H_N / TN, B_N / TM, E_N), 256, 0, stream>>>(
      zp, glist, counts, offsets, cW1, cb1, cg, cbt, cm, cv, hbuf /* as h2g */);
  k6_expert2<<<dim3(Z_N / TN, B_N / TM, E_N), 256, 0, stream>>>(
      hbuf /* h2g */, counts, offsets, cW2, cb2, z2raw);
  k7_final<<<B_N / 8, 256, 0, stream>>>(
      z2raw, glist, counts, offsets, cProto, proj_class, logits_class);
}